// DEMOWeightLayer_3083786518799
// MI455X (gfx1250) — compile-verified
//
#include <hip/hip_runtime.h>

typedef __attribute__((ext_vector_type(2))) float v2f;
typedef __attribute__((ext_vector_type(8))) float v8f;

#define IN_DIM 64
#define OUT_DIM 64
#define LDS_STRIDE 68   // padded row stride (floats): conflict-free b64 reads

// D = A(16x4) * B(4x16) + C, all fp32, full precision.
#define WMMA_F32_16x16x4(A, B, C) \
  __builtin_amdgcn_wmma_f32_16x16x4_f32(false, (A), false, (B), (short)0, (C), false, false)

// ---------------------------------------------------------------------------
// Phase 1: edge scatter. 16 threads per edge; each moves 4 floats of x[src]
// into neigh_sum[dst] with f32 atomics. Lane 0 of each edge counts degree.
// (102M f32 atomics, all hitting the 192MB L2 -> this is the runtime floor.)
// ---------------------------------------------------------------------------
__global__ __launch_bounds__(256) void demo_scatter_kernel(
    const float* __restrict__ x, const int* __restrict__ ei,
    float* __restrict__ nsum, float* __restrict__ deg, int n_edges) {
  long long tid = (long long)blockIdx.x * blockDim.x + threadIdx.x;
  int e = (int)(tid >> 4);
  if (e >= n_edges) return;
  int part = (int)(tid & 15);
  int s = ei[e];              // edge_index[0][e]
  int d = ei[n_edges + e];    // edge_index[1][e]
  const float4 v = *(const float4*)(x + (size_t)s * IN_DIM + part * 4);
  float* dst = nsum + (size_t)d * IN_DIM + part * 4;
  atomicAdd(dst + 0, v.x);
  atomicAdd(dst + 1, v.y);
  atomicAdd(dst + 2, v.z);
  atomicAdd(dst + 3, v.w);
  if (part == 0) atomicAdd(deg + d, 1.0f);
}

// ---------------------------------------------------------------------------
// Phase 2: per-node GEMMs with V_WMMA_F32_16X16X4_F32.
// One wave32 per 16 nodes x 64 output dims. Weights staged in LDS (padded
// stride, bank-conflict-free ds_load_b64 fragments).
//   out = x @ Wg^T + (hn * x) @ Ws^T + (nsum/max(deg,1)) @ Wl^T + bias, ELU
// ---------------------------------------------------------------------------
__global__ __launch_bounds__(256) void demo_node_kernel(
    const float* __restrict__ x, const float* __restrict__ Wg,
    const float* __restrict__ Wl, const float* __restrict__ Ws,
    const float* __restrict__ bias, const float* __restrict__ deg,
    float* sum_out, int n_nodes) {
  __shared__ float sW[3 * OUT_DIM * LDS_STRIDE];  // 3 x 64 x 68 floats ~ 51 KB
  float* sWg = sW;
  float* sWs = sW + OUT_DIM * LDS_STRIDE;
  float* sWl = sW + 2 * OUT_DIM * LDS_STRIDE;

  // ---- stage weights: 1024 float4 per matrix, 256 threads -> 4 each -------
  {
    const int tid = threadIdx.x;
#pragma unroll
    for (int i = 0; i < 4; ++i) {
      int idx = tid + i * 256;            // float4 index 0..1023
      int row = idx >> 4;                 // 0..63
      int k4  = idx & 15;                 // 0..15 (float4 within row)
      float4 g = *(const float4*)(Wg + (size_t)idx * 4);
      float4 s = *(const float4*)(Ws + (size_t)idx * 4);
      float4 l = *(const float4*)(Wl + (size_t)idx * 4);
      *(float4*)(sWg + row * LDS_STRIDE + k4 * 4) = g;
      *(float4*)(sWs + row * LDS_STRIDE + k4 * 4) = s;
      *(float4*)(sWl + row * LDS_STRIDE + k4 * 4) = l;
    }
  }
  __syncthreads();

  const int lane    = threadIdx.x & 31;
  const int wave    = threadIdx.x >> 5;
  const int lane_lo = lane & 15;
  const int hi      = lane >> 4;   // 0: K=0..1, 1: K=2..3 within a chunk
  const int node_base = (blockIdx.x * 8 + wave) * 16;
  if (node_base >= n_nodes) return;   // wave-uniform exit (after barrier)

  // A-fragment row; tail clamps stay inside this wave's own 16 rows.
  int arow = node_base + lane_lo;
  if (arow > n_nodes - 1) arow = n_nodes - 1;

  const float dv  = deg[arow];
  const float inv = 1.0f / fmaxf(dv, 1.0f);        // mean divisor
  const float hn  = dv > 0.0f ? 1.0f : 0.0f;       // mask for the Ws term

  // ISA 16x4 f32 A layout: lane L -> M = L&15, VGPRs hold K = 2*hi + {0,1}.
  const v2f* xr = (const v2f*)(x       + (size_t)arow * IN_DIM + 2 * hi);
  const v2f* sr = (const v2f*)(sum_out + (size_t)arow * IN_DIM + 2 * hi);

  // B-fragment LDS base for this lane: row = nb*16 + lane_lo, col = 2*hi+4c.
  const int bofs = lane_lo * LDS_STRIDE + 2 * hi;

  v8f acc[4] = {v8f{}, v8f{}, v8f{}, v8f{}};

  for (int c = 0; c < 16; ++c) {          // K chunks of 4 (K total = 64)
    v2f ax = xr[2 * c];                   // x fragment
    v2f am = sr[2 * c] * inv;             // neigh_mean fragment
    v2f as = ax * hn;                     // masked x fragment for Ws

    // Fetch all 12 B fragments of this chunk first (one DS clause)...
    v2f bg[4], bs[4], bl[4];
#pragma unroll
    for (int nb = 0; nb < 4; ++nb) {
      const int bo = bofs + nb * 16 * LDS_STRIDE + 4 * c;
      bg[nb] = *(const v2f*)(sWg + bo);
      bs[nb] = *(const v2f*)(sWs + bo);
      bl[nb] = *(const v2f*)(sWl + bo);
    }
    // ...then issue the 12 WMMAs back-to-back.
#pragma unroll
    for (int nb = 0; nb < 4; ++nb) {
      acc[nb] = WMMA_F32_16x16x4(ax, bg[nb], acc[nb]);
      acc[nb] = WMMA_F32_16x16x4(as, bs[nb], acc[nb]);
      acc[nb] = WMMA_F32_16x16x4(am, bl[nb], acc[nb]);
    }
  }

  // Epilogue: bias + ELU, store (D layout: VGPR r -> M = r + 8*hi, N = lane&15)
#pragma unroll
  for (int nb = 0; nb < 4; ++nb) {
    const int col = nb * 16 + lane_lo;
    const float bv = bias[col];
    v8f r = acc[nb];
#pragma unroll
    for (int rr = 0; rr < 8; ++rr) {
      int m = node_base + rr + 8 * hi;
      if (m < n_nodes) {
        float val = r[rr] + bv;
        val = val > 0.0f ? val : (__expf(val) - 1.0f);
        sum_out[(size_t)m * OUT_DIM + col] = val;
      }
    }
  }
}

// ---------------------------------------------------------------------------
extern "C" void kernel_launch(void* const* d_in, const int* in_sizes, int n_in,
                              void* d_out, int out_size, void* d_ws, size_t ws_size,
                              hipStream_t stream) {
  const float* x    = (const float*)d_in[0];
  const float* Wg   = (const float*)d_in[1];
  const float* Wl   = (const float*)d_in[2];
  const float* Ws   = (const float*)d_in[3];
  const float* bias = (const float*)d_in[4];
  const int*   ei   = (const int*)d_in[5];   // JAX x64-off: int32, shape [2,E]
  float* out = (float*)d_out;                // doubles as neigh_sum accumulator
  float* deg = (float*)d_ws;                 // N floats of scratch

  const int n_nodes = in_sizes[0] / IN_DIM;
  const int n_edges = in_sizes[5] / 2;

  // Zero accumulators every call (harness poisons buffers, graph replays).
  hipMemsetAsync(out, 0, (size_t)n_nodes * IN_DIM * sizeof(float), stream);
  hipMemsetAsync(deg, 0, (size_t)n_nodes * sizeof(float), stream);

  {
    long long threads = (long long)n_edges * 16;
    int blocks = (int)((threads + 255) / 256);
    demo_scatter_kernel<<<blocks, 256, 0, stream>>>(x, ei, out, deg, n_edges);
  }
  {
    int waves  = (n_nodes + 15) / 16;
    int blocks = (waves + 7) / 8;   // 8 waves (256 threads) per block
    demo_node_kernel<<<blocks, 256, 0, stream>>>(x, Wg, Wl, Ws, bias, deg, out,
                                                 n_nodes);
  }
}